// LocalContrastNormalization_56298431316297
// MI455X (gfx1250) — compile-verified
//
#include <hip/hip_runtime.h>
#include <hip/hip_bf16.h>

typedef __attribute__((ext_vector_type(2))) float v2f;
typedef __attribute__((ext_vector_type(8))) float v8f;

#define TW 128          // tile width  (output cols per WG)
#define TH 64           // tile height (output rows per WG)
#define HALO_ROWS 94    // TH + 30
#define XT_W 160        // loaded cols: col0 = 128*bx - 16, 40 x B128 per row
#define PITCH 164       // LDS row pitch in floats (4-float pad; 164*4 % 16 == 0)
#define W 1024
#define H 1024
#define K2INV (1.0f / 961.0f)
#define EPS 1e-5f

// LDS layout (dynamic): xt[94*164] | vs[64*164] | vs2[64*164]  = 145,632 B
#define XT_ELEMS (HALO_ROWS * PITCH)
#define VS_ELEMS (TH * PITCH)
#define SHMEM_BYTES ((XT_ELEMS + 2 * VS_ELEMS) * 4)

__global__ void __launch_bounds__(256)
LocalContrastNormalization_56298431316297_kernel(const float* __restrict__ x,
                                                 float* __restrict__ out) {
  extern __shared__ float smem[];
  float* xt  = smem;                 // halo input tile
  float* vs  = smem + XT_ELEMS;      // vertical box sums of x
  float* vs2 = vs + VS_ELEMS;        // vertical box sums of x^2

  const int t  = threadIdx.x;
  const int bx = blockIdx.x;   // 0..7   (cols of 128)
  const int by = blockIdx.y;   // 0..15  (rows of 64)
  const int bz = blockIdx.z;   // batch

  const int row0 = TH * by - 15;     // global row of xt row 0
  const int col0 = TW * bx - 16;     // global col of xt col 0 (16-float aligned)
  const float* src = x + (size_t)bz * (size_t)(H * W);

  // ---- border tiles: zero-fill halo tile first (zero padding semantics) ----
  if (bx == 0 || bx == 7 || by == 0 || by == 15) {
    float4* p = (float4*)xt;
    float4 z; z.x = z.y = z.z = z.w = 0.0f;
    for (int i = t; i < XT_ELEMS / 4; i += 256) p[i] = z;
    __syncthreads();  // DS zero-stores complete before async-to-LDS writes
  }

  // ---- async global -> LDS tile load (B128 per lane), predicated in-bounds ----
  for (int o = t; o < HALO_ROWS * 40; o += 256) {
    int r  = o / 40;
    int s4 = o - 40 * r;            // 16-byte segment within row
    int grow = row0 + r;
    int gcol = col0 + 4 * s4;
    if ((unsigned)grow < (unsigned)H && (unsigned)gcol <= (unsigned)(W - 4)) {
      unsigned ldsA = (unsigned)(size_t)&xt[r * PITCH + 4 * s4];
      unsigned voff = (unsigned)((grow * W + gcol) * 4);
      asm volatile("global_load_async_to_lds_b128 %0, %1, %2"
                   :: "v"(ldsA), "v"(voff), "s"(src)
                   : "memory");
    }
  }
  asm volatile("s_wait_asynccnt 0" ::: "memory");
  __syncthreads();

  // ---- vertical 31-tap box sums (sliding window per column) ----
  if (t < XT_W) {
    float s = 0.0f, s2 = 0.0f;
#pragma unroll
    for (int i = 0; i < 31; ++i) {
      float v = xt[i * PITCH + t];
      s += v;
      s2 = fmaf(v, v, s2);
    }
    for (int r = 0; r < TH; ++r) {
      vs[r * PITCH + t]  = s;
      vs2[r * PITCH + t] = s2;
      if (r < TH - 1) {
        float vn = xt[(r + 31) * PITCH + t];
        float vo = xt[r * PITCH + t];
        s  += vn - vo;
        s2 += fmaf(vn, vn, -vo * vo);
      }
    }
  } else if (t < PITCH) {
    // zero the pitch-pad columns (read by the K=46..47 pad chunks below)
    for (int r = 0; r < TH; ++r) {
      vs[r * PITCH + t]  = 0.0f;
      vs2[r * PITCH + t] = 0.0f;
    }
  }
  __syncthreads();

  // ---- horizontal 31-tap box sums as banded matmul on the FP32 matrix core ----
  // out16x16 = A(16x48) x S(48x16),  A[m][k] = vsum[r0+m][c0+1+k],
  // S[k][n] = 1 iff n <= k <= n+30  (zero in the K=46..47 pad region).
  const unsigned lane = (unsigned)t & 31u;
  const unsigned wid  = (unsigned)t >> 5;
  const int m   = (int)(lane & 15u);   // A: M row / B: N col / D: N col
  const int grp = (int)(lane >> 4);    // K pair select

  v2f bf[12];
#pragma unroll
  for (int kc = 0; kc < 12; ++kc) {
    int k0 = 4 * kc + 2 * grp;
    bf[kc].x = (k0 >= m && k0 <= m + 30) ? 1.0f : 0.0f;
    bf[kc].y = (k0 + 1 >= m && k0 + 1 <= m + 30) ? 1.0f : 0.0f;
  }

  float* outBase = out + (size_t)bz * (size_t)(H * W);

  for (int jj = 0; jj < 4; ++jj) {           // 32 tiles / 8 waves
    int job = (int)wid + 8 * jj;
    int r0 = (job >> 3) * 16;
    int c0 = (job & 7) * 16;

    v8f acc1 = {0.f, 0.f, 0.f, 0.f, 0.f, 0.f, 0.f, 0.f};
    v8f acc2 = {0.f, 0.f, 0.f, 0.f, 0.f, 0.f, 0.f, 0.f};

    const float* A1 = vs  + (r0 + m) * PITCH + (c0 + 1);
    const float* A2 = vs2 + (r0 + m) * PITCH + (c0 + 1);

#pragma unroll
    for (int kc = 0; kc < 12; ++kc) {
      int kb = 4 * kc + 2 * grp;
      v2f a1 = { A1[kb], A1[kb + 1] };
      v2f a2 = { A2[kb], A2[kb + 1] };
      acc1 = __builtin_amdgcn_wmma_f32_16x16x4_f32(false, a1, false, bf[kc],
                                                   (short)0, acc1, false, false);
      acc2 = __builtin_amdgcn_wmma_f32_16x16x4_f32(false, a2, false, bf[kc],
                                                   (short)0, acc2, false, false);
    }

    // ---- pointwise LCN + store (D layout: lanes 0-15 -> M=v, 16-31 -> M=v+8) ----
#pragma unroll
    for (int v = 0; v < 8; ++v) {
      int r = r0 + v + 8 * grp;         // output row within tile block
      int c = c0 + m;                   // output col within tile block
      float hs   = acc1[v];
      float hq   = acc2[v];
      float mean = hs * K2INV;
      float var  = fmaf(-mean, mean, hq * K2INV);
      float sd   = sqrtf(fmaxf(var, EPS));
      float xv   = xt[(r + 15) * PITCH + (c + 16)];
      float nrm  = (xv - mean) / (sd + EPS);
      float enh  = 1.0f / (1.0f + __expf(-0.5f * nrm));
      outBase[(size_t)(TH * by + r) * W + (TW * bx + c)] = enh;
    }
  }
}

extern "C" void kernel_launch(void* const* d_in, const int* in_sizes, int n_in,
                              void* d_out, int out_size, void* d_ws, size_t ws_size,
                              hipStream_t stream) {
  (void)in_sizes; (void)n_in; (void)d_ws; (void)ws_size; (void)out_size;
  const float* x = (const float*)d_in[0];
  float* out = (float*)d_out;

  (void)hipFuncSetAttribute(
      reinterpret_cast<const void*>(&LocalContrastNormalization_56298431316297_kernel),
      hipFuncAttributeMaxDynamicSharedMemorySize, (int)SHMEM_BYTES);

  dim3 grid(W / TW, H / TH, 32);   // 8 x 16 x 32
  dim3 block(256);
  LocalContrastNormalization_56298431316297_kernel<<<grid, block, SHMEM_BYTES, stream>>>(x, out);
}